// VQQuantizer_24129126269383
// MI455X (gfx1250) — compile-verified
//
#include <hip/hip_runtime.h>
#include <hip/hip_bf16.h>

typedef __attribute__((ext_vector_type(2))) float v2f;
typedef __attribute__((ext_vector_type(8))) float v8f;

#define HID   256
#define NTOK  8192
#define NBAT  8192
#define TILE  16            // token tile width (WMMA N)
#define BPAD  260           // padded LDS row stride in floats (260 % 64 == 4 -> conflict-free frag reads)
#define WVWG  4             // waves per workgroup
#define ROWS  (16 * WVWG)   // rows of h per workgroup
#define NTILES (NTOK / TILE)

// ---------------- zero the one-hot region (256 MB) ----------------
__global__ void vq_zero_kernel(float4* __restrict__ p, long n4) {
    long i = (long)blockIdx.x * blockDim.x + threadIdx.x;
    long stride = (long)gridDim.x * blockDim.x;
    float4 z = make_float4(0.f, 0.f, 0.f, 0.f);
    for (; i < n4; i += stride) p[i] = z;
}

// ---------------- c_sq[k] = sum_h codebook[k][h]^2, one wave32 per row ----------------
__global__ __launch_bounds__(256) void vq_csq_kernel(const float* __restrict__ cb,
                                                     float* __restrict__ csq) {
    int wave = blockIdx.x * (blockDim.x >> 5) + (threadIdx.x >> 5);
    int lane = threadIdx.x & 31;
    const float* row = cb + (size_t)wave * HID + lane * 8;
    float4 a = *(const float4*)(row);
    float4 b = *(const float4*)(row + 4);
    float s = a.x*a.x + a.y*a.y + a.z*a.z + a.w*a.w
            + b.x*b.x + b.y*b.y + b.z*b.z + b.w*b.w;
    #pragma unroll
    for (int m = 16; m >= 1; m >>= 1) s += __shfl_xor(s, m);
    if (lane == 0) csq[wave] = s;
}

// ---------------- main: f32 WMMA distance + running argmin, async double-buffered B ----------------
__global__ __launch_bounds__(WVWG * 32) void vq_argmin_kernel(
        const float* __restrict__ h, const float* __restrict__ cb,
        const float* __restrict__ csq, int* __restrict__ idx_out) {
    __shared__ float Bt[2 * TILE * BPAD];   // double-buffered 16x256 codebook tile (padded)

    const int t    = threadIdx.x;
    const int lane = t & 31;
    const int wave = t >> 5;
    const int lm   = lane & 15;   // M row (A) / N col (B) within fragment
    const int lh   = lane >> 4;   // selects K-pair {0,1} vs {2,3}
    const int rowBase = blockIdx.x * ROWS + wave * 16;

    // A fragments for the whole H=256 reduction live in registers: 64 x v2f.
    // ISA 32-bit A 16x4 layout: lane L -> M = L%16, VGPR pair = K {0,1} (L<16) or {2,3} (L>=16).
    v2f a[HID / 4];
    {
        const float* hrow = h + (size_t)(rowBase + lm) * HID + lh * 2;
        #pragma unroll
        for (int kk = 0; kk < HID / 4; ++kk)
            a[kk] = *(const v2f*)(hrow + kk * 4);
    }

    // --- async staging setup: 128 threads cover 16 rows x 256 floats; 8 x b128 per thread ---
    const int r  = t >> 3;          // token row within tile (0..15), 8 threads per row
    const int c0 = (t & 7) * 32;    // 32-float span per thread
    const unsigned lds_base = (unsigned)(uintptr_t)(&Bt[0]);           // LDS byte offset of Bt
    const unsigned lds_toff = (unsigned)((r * BPAD + c0) * 4);         // this thread's slot
    const unsigned long long sb = (unsigned long long)(uintptr_t)cb;   // SGPR base for GVS mode

    auto stage = [&](int buf, int tileIdx) {
        unsigned ldsv = lds_base + (unsigned)buf * (TILE * BPAD * 4u) + lds_toff;
        unsigned voff = ((unsigned)(tileIdx * TILE + r) << 10) + ((unsigned)(t & 7) << 7);
        // INST_OFFSET is added to BOTH the global and LDS address (ISA 08 §4.4),
        // and our per-chunk delta is 16 B on both sides -> one (lds, vaddr, saddr) triple.
        asm volatile(
            "global_load_async_to_lds_b128 %0, %1, %2 offset:0\n\t"
            "global_load_async_to_lds_b128 %0, %1, %2 offset:16\n\t"
            "global_load_async_to_lds_b128 %0, %1, %2 offset:32\n\t"
            "global_load_async_to_lds_b128 %0, %1, %2 offset:48\n\t"
            "global_load_async_to_lds_b128 %0, %1, %2 offset:64\n\t"
            "global_load_async_to_lds_b128 %0, %1, %2 offset:80\n\t"
            "global_load_async_to_lds_b128 %0, %1, %2 offset:96\n\t"
            "global_load_async_to_lds_b128 %0, %1, %2 offset:112"
            :: "v"(ldsv), "v"(voff), "s"(sb) : "memory");
    };

    float minv[8];
    int   mini[8];
    #pragma unroll
    for (int v = 0; v < 8; ++v) { minv[v] = 3.4e38f; mini[v] = 0; }

    stage(0, 0);   // prime the pipeline (ASYNCcnt = 8 per wave)

    for (int kt = 0; kt < NTILES; ++kt) {
        const int cur = kt & 1;
        const int tokBase = kt * TILE;

        // prefetch next tile into the other buffer (dummy re-fetch of tile 0 on last iter
        // keeps the wait immediate constant); previous batch completes in order.
        const int nt = (kt + 1 < NTILES) ? kt + 1 : 0;
        stage(cur ^ 1, nt);
        asm volatile("s_wait_asynccnt 0x8" ::: "memory");  // current tile's 8 loads landed
        __syncthreads();                                   // publish across waves

        const float cs = csq[tokBase + lm];

        // 64 chained f32 WMMAs accumulate the full 16x16 dot-product tile
        v8f acc = (v8f){0.f, 0.f, 0.f, 0.f, 0.f, 0.f, 0.f, 0.f};
        const float* bb = &Bt[cur * TILE * BPAD + lm * BPAD + lh * 2];  // B 4x16 layout mirrors A
        #pragma unroll
        for (int kk = 0; kk < HID / 4; ++kk) {
            v2f b = *(const v2f*)(bb + kk * 4);
            acc = __builtin_amdgcn_wmma_f32_16x16x4_f32(
                /*neg_a=*/false, a[kk], /*neg_b=*/false, b,
                /*c_mod=*/(short)0, acc, /*reuse_a=*/false, /*reuse_b=*/false);
        }

        // D layout: VGPR v holds row v (lanes 0-15) or row v+8 (lanes 16-31), col = lane%16
        const int tok = tokBase + lm;
        #pragma unroll
        for (int v = 0; v < 8; ++v) {
            float d = cs - 2.0f * acc[v];   // h_sq omitted: constant per row, argmin-invariant
            if (d < minv[v]) { minv[v] = d; mini[v] = tok; }
        }
        __syncthreads();   // all waves done reading Bt[cur] before it is overwritten next iter
    }

    // argmin reduce across the 16 lanes sharing each row set (first-index tie-break)
    #pragma unroll
    for (int off = 8; off >= 1; off >>= 1) {
        #pragma unroll
        for (int v = 0; v < 8; ++v) {
            float ov = __shfl_xor(minv[v], off);
            int   oi = __shfl_xor(mini[v], off);
            if (ov < minv[v] || (ov == minv[v] && oi < mini[v])) {
                minv[v] = ov; mini[v] = oi;
            }
        }
    }
    if (lm == 0) {
        #pragma unroll
        for (int v = 0; v < 8; ++v)
            idx_out[rowBase + lh * 8 + v] = mini[v];
    }
}

// ---------------- loss + one-hot scatter, one wave32 per row ----------------
__global__ __launch_bounds__(256) void vq_loss_kernel(
        const float* __restrict__ h, const float* __restrict__ cb,
        const int* __restrict__ idx, float* __restrict__ hard,
        float* __restrict__ loss) {
    int n    = blockIdx.x * (blockDim.x >> 5) + (threadIdx.x >> 5);
    int lane = threadIdx.x & 31;
    int id   = idx[n];
    const float* hr = h  + (size_t)n  * HID + lane * 8;
    const float* cr = cb + (size_t)id * HID + lane * 8;
    float s = 0.f;
    #pragma unroll
    for (int j = 0; j < 8; ++j) { float d = hr[j] - cr[j]; s += d * d; }
    #pragma unroll
    for (int m = 16; m >= 1; m >>= 1) s += __shfl_xor(s, m);
    if (lane == 0) {
        // codebook_loss + 0.25*commit_loss, both numerically mean((h - z_q)^2)
        loss[n] = 1.25f * s * (1.0f / (float)HID);
        hard[(size_t)n * NTOK + id] = 1.0f;
    }
}

extern "C" void kernel_launch(void* const* d_in, const int* in_sizes, int n_in,
                              void* d_out, int out_size, void* d_ws, size_t ws_size,
                              hipStream_t stream) {
    const float* h  = (const float*)d_in[0];
    // d_in[1] = temperature (unused, API compat)
    const float* cb = (const float*)d_in[2];

    float* hard = (float*)d_out;                         // (N, K) one-hot
    float* loss = (float*)d_out + (size_t)NBAT * NTOK;   // (N,)  loss

    int*   idx = (int*)d_ws;                                 // N ints
    float* csq = (float*)((char*)d_ws + NBAT * sizeof(int)); // K floats

    // 1) zero the 256 MB one-hot region
    vq_zero_kernel<<<2048, 256, 0, stream>>>((float4*)hard,
                                             (long)((size_t)NBAT * NTOK / 4));
    // 2) per-token squared norms
    vq_csq_kernel<<<NTOK / 8, 256, 0, stream>>>(cb, csq);
    // 3) WMMA-f32 distance GEMM + running argmin (no (N,K) materialization)
    vq_argmin_kernel<<<NBAT / ROWS, WVWG * 32, 0, stream>>>(h, cb, csq, idx);
    // 4) loss + scatter ones
    vq_loss_kernel<<<NBAT / 8, 256, 0, stream>>>(h, cb, idx, hard, loss);
}